// SageBlock_24816321036489
// MI455X (gfx1250) — compile-verified
//
#include <hip/hip_runtime.h>

#define D 128
#define LDSS 132  // padded LDS row stride (floats) to avoid 64-bank conflicts

typedef __attribute__((ext_vector_type(2))) float v2f;
typedef __attribute__((ext_vector_type(8))) float v8f;

// ---------------------------------------------------------------------------
// CSR build step 1: per-destination degree histogram (int atomics).
// ---------------------------------------------------------------------------
__global__ __launch_bounds__(256) void csr_count(
    const long long* __restrict__ ei, int* __restrict__ deg, int E)
{
    int e = blockIdx.x * 256 + threadIdx.x;
    if (e < E) atomicAdd(&deg[(int)ei[(size_t)E + e]], 1);
}

// ---------------------------------------------------------------------------
// CSR build step 2: single-block exclusive scan, in place (deg -> offsets).
// Also primes the fill cursors. N <= 1024 * chunk.
// ---------------------------------------------------------------------------
__global__ __launch_bounds__(1024) void csr_exscan(
    int* __restrict__ deg, int* __restrict__ cursor, int N)
{
    __shared__ int part[1024];
    const int t = threadIdx.x;
    const int C = (N + 1023) / 1024;
    const int lo = t * C;
    const int hi = min(lo + C, N);

    int s = 0;
    for (int i = lo; i < hi; ++i) s += deg[i];
    part[t] = s;
    __syncthreads();
    for (int off = 1; off < 1024; off <<= 1) {   // Hillis-Steele inclusive scan
        int v = (t >= off) ? part[t - off] : 0;
        __syncthreads();
        part[t] += v;
        __syncthreads();
    }
    int prefix = (t == 0) ? 0 : part[t - 1];
    for (int i = lo; i < hi; ++i) {
        int d = deg[i];
        deg[i] = prefix;      // exclusive offset
        cursor[i] = prefix;   // fill cursor
        prefix += d;
    }
    if (t == 0) deg[N] = part[1023];             // total edge count
}

// ---------------------------------------------------------------------------
// CSR build step 3: bin source ids by destination.
// ---------------------------------------------------------------------------
__global__ __launch_bounds__(256) void csr_fill(
    const long long* __restrict__ ei, int* __restrict__ cursor,
    int* __restrict__ ssrc, int E)
{
    int e = blockIdx.x * 256 + threadIdx.x;
    if (e < E) {
        int d = (int)ei[(size_t)E + e];
        int s = (int)ei[e];
        int p = atomicAdd(&cursor[d], 1);
        ssrc[p] = s;
    }
}

// ---------------------------------------------------------------------------
// Fused layer: pull-mode mean aggregation (no atomics, x is L2-resident) +
// out = agg@Wl^T + bl + x@Wr^T via f32 WMMA 16x16x4 + LayerNorm + ReLU.
// One 16-node tile per 128-thread block (4 waves); each wave: 2 col tiles.
// ---------------------------------------------------------------------------
__global__ __launch_bounds__(128) void sage_fused(
    const float* __restrict__ xin,
    const int* __restrict__ offsets, const int* __restrict__ ssrc,
    const float* __restrict__ Wl, const float* __restrict__ bl,
    const float* __restrict__ Wr,
    const float* __restrict__ gamma, const float* __restrict__ beta,
    float* __restrict__ out, int N)
{
    __shared__ float sh_agg[16 * LDSS];
    __shared__ float sh_x[16 * LDSS];
    __shared__ float sh_o[16 * LDSS];

    const int tid  = threadIdx.x;
    const int m0   = blockIdx.x * 16;
    const int lane = tid & 31;
    const int wv   = tid >> 5;

    // Stage 16x128 x-tile into LDS (coalesced float4).
    for (int i = tid; i < 512; i += 128) {
        int r = i >> 5, c4 = i & 31, gr = m0 + r;
        float4 xv = make_float4(0.f, 0.f, 0.f, 0.f);
        if (gr < N) xv = ((const float4*)(xin + (size_t)gr * D))[c4];
        float* px = sh_x + r * LDSS + c4 * 4;
        px[0] = xv.x; px[1] = xv.y; px[2] = xv.z; px[3] = xv.w;
    }

    // Pull-mode mean aggregation: wave wv owns rows wv*4 .. wv*4+3.
    // Neighbor ids loaded coalesced once per wave, broadcast via shuffle;
    // each lane accumulates one float4 (32 lanes x float4 = 128 cols).
    for (int rr = 0; rr < 4; ++rr) {
        int r = wv * 4 + rr, gr = m0 + r;
        float4 acc = make_float4(0.f, 0.f, 0.f, 0.f);
        int off0 = 0, off1 = 0;
        if (gr < N) { off0 = offsets[gr]; off1 = offsets[gr + 1]; }
        for (int base = off0; base < off1; base += 32) {
            int nn  = min(32, off1 - base);
            int idx = (base + lane < off1) ? ssrc[base + lane] : 0;
            for (int j = 0; j < nn; ++j) {
                int s = __shfl(idx, j, 32);
                float4 v = ((const float4*)(xin + (size_t)s * D))[lane];
                acc.x += v.x; acc.y += v.y; acc.z += v.z; acc.w += v.w;
            }
        }
        float inv = 1.0f / fmaxf((float)(off1 - off0), 1.0f);
        float* pa = sh_agg + r * LDSS + lane * 4;
        pa[0] = acc.x * inv; pa[1] = acc.y * inv;
        pa[2] = acc.z * inv; pa[3] = acc.w * inv;
    }
    __syncthreads();

    const int row = lane & 15;        // A: M index / B: N index
    const int kh  = (lane >> 4) * 2;  // lanes 16-31 hold K+2/K+3

    for (int t = 0; t < 2; ++t) {
        const int n0 = (wv * 2 + t) * 16;
        v8f acc = {};
        const float* wlrow = Wl + (size_t)(n0 + row) * D;  // B[k][n] = Wl[n][k]
        const float* wrrow = Wr + (size_t)(n0 + row) * D;
        const float* arow  = sh_agg + row * LDSS;
        const float* xrow  = sh_x   + row * LDSS;
#pragma unroll
        for (int kt = 0; kt < 32; ++kt) {        // agg @ Wl^T
            int kb = kt * 4 + kh;
            v2f a, b;
            a.x = arow[kb];  a.y = arow[kb + 1];
            b.x = wlrow[kb]; b.y = wlrow[kb + 1];
            acc = __builtin_amdgcn_wmma_f32_16x16x4_f32(
                false, a, false, b, (short)0, acc, false, false);
        }
#pragma unroll
        for (int kt = 0; kt < 32; ++kt) {        // + x @ Wr^T (same accumulator)
            int kb = kt * 4 + kh;
            v2f a, b;
            a.x = xrow[kb];  a.y = xrow[kb + 1];
            b.x = wrrow[kb]; b.y = wrrow[kb + 1];
            acc = __builtin_amdgcn_wmma_f32_16x16x4_f32(
                false, a, false, b, (short)0, acc, false, false);
        }
        // C/D layout: VGPR r -> M = r + (lane>=16 ? 8 : 0), N = lane&15
        const float bias = bl[n0 + row];
        const int   mo   = (lane >> 4) * 8;
#pragma unroll
        for (int r = 0; r < 8; ++r)
            sh_o[(r + mo) * LDSS + n0 + row] = acc[r] + bias;
    }
    __syncthreads();

    // LayerNorm + ReLU: 8 threads per node row, shuffle-reduce within wave32.
    {
        int r   = tid >> 3;   // 0..15
        int sub = tid & 7;
        const float* orow = sh_o + r * LDSS;
        float s1 = 0.f, s2 = 0.f;
#pragma unroll
        for (int c = sub * 16; c < sub * 16 + 16; ++c) {
            float v = orow[c];
            s1 += v; s2 += v * v;
        }
#pragma unroll
        for (int m = 1; m < 8; m <<= 1) {
            s1 += __shfl_xor(s1, m, 32);
            s2 += __shfl_xor(s2, m, 32);
        }
        float mu  = s1 * (1.0f / D);
        float var = s2 * (1.0f / D) - mu * mu;
        float rn  = rsqrtf(var + 1e-5f);
        int gr = m0 + r;
        if (gr < N) {
            float* po = out + (size_t)gr * D;
#pragma unroll
            for (int c = sub * 16; c < sub * 16 + 16; ++c) {
                float v = (orow[c] - mu) * rn * gamma[c] + beta[c];
                po[c] = fmaxf(v, 0.0f);
            }
        }
    }
}

// ---------------------------------------------------------------------------
extern "C" void kernel_launch(void* const* d_in, const int* in_sizes, int n_in,
                              void* d_out, int out_size, void* d_ws, size_t ws_size,
                              hipStream_t stream)
{
    const float*     x  = (const float*)d_in[0];
    const long long* ei = (const long long*)d_in[1];   // int64 in reference
    const float* Wl0 = (const float*)d_in[2];
    const float* bl0 = (const float*)d_in[3];
    const float* Wr0 = (const float*)d_in[4];
    const float* g0  = (const float*)d_in[5];
    const float* be0 = (const float*)d_in[6];
    const float* Wl1 = (const float*)d_in[7];
    const float* bl1 = (const float*)d_in[8];
    const float* Wr1 = (const float*)d_in[9];
    const float* g1  = (const float*)d_in[10];
    const float* be1 = (const float*)d_in[11];

    const int N = in_sizes[0] / D;
    const int E = in_sizes[1] / 2;

    // Workspace: offsets[N+1] | cursor[N] | ssrc[E] | (16B-align) | y[N*D]
    int* offsets = (int*)d_ws;
    int* cursor  = offsets + (N + 1);
    int* ssrc    = cursor + N;
    size_t felts = (size_t)(N + 1) + N + E;
    felts = (felts + 3) & ~(size_t)3;              // 16-byte align y
    float* y = (float*)d_ws + felts;

    dim3 eblk(256), egrd((unsigned)((E + 255) / 256));
    dim3 gblk(128), ggrd((unsigned)((N + 15) / 16));

    // ---- build CSR once (shared by both layers) ----
    hipMemsetAsync(offsets, 0, (size_t)(N + 1) * sizeof(int), stream);
    csr_count <<<egrd, eblk, 0, stream>>>(ei, offsets, E);
    csr_exscan<<<1, 1024, 0, stream>>>(offsets, cursor, N);
    csr_fill  <<<egrd, eblk, 0, stream>>>(ei, cursor, ssrc, E);

    // ---- layer 0 ----
    sage_fused<<<ggrd, gblk, 0, stream>>>(x, offsets, ssrc,
                                          Wl0, bl0, Wr0, g0, be0, y, N);
    // ---- layer 1 ----
    sage_fused<<<ggrd, gblk, 0, stream>>>(y, offsets, ssrc,
                                          Wl1, bl1, Wr1, g1, be1,
                                          (float*)d_out, N);
}